// FootprintExtruder_91216515432457
// MI455X (gfx1250) — compile-verified
//
#include <hip/hip_runtime.h>

typedef int v4i __attribute__((ext_vector_type(4)));

#define H_DIM   768
#define W_DIM   768
#define HW      (H_DIM * W_DIM)     // 589824 pixels per z-slice
#define ZMAX    384
#define Z_CHUNK 16                  // z-slices per block (384/16 = 24 chunks)
#define FP_MIN  100
#define FP_MAX  5000
#define THREADS 256                 // 8 wave32 per block

// Each block handles 1024 contiguous pixels (256 threads x int4) for Z_CHUNK
// z-slices. seg/hf tiles are staged global->LDS with the gfx1250 async path,
// then the z-loop issues one non-temporal b128 store per slice per thread.
__global__ __launch_bounds__(THREADS) void footprint_extrude_kernel(
    const int* __restrict__ height_field,
    const int* __restrict__ seg_map,
    int* __restrict__ out)
{
    __shared__ v4i s_tile[2 * THREADS];   // [0..255] = seg4, [256..511] = hf4

    const int      tid     = threadIdx.x;
    const unsigned byteOff = blockIdx.x * (THREADS * 16u) + (unsigned)tid * 16u;

    // LDS byte offsets: low 32 bits of a flat LDS pointer are the LDS offset.
    const unsigned lds_seg = (unsigned)(unsigned long long)(&s_tile[tid]);
    const unsigned lds_hf  = (unsigned)(unsigned long long)(&s_tile[THREADS + tid]);

    // Async DMA global -> LDS (per-lane 16B), tracked by ASYNCcnt.
    asm volatile("global_load_async_to_lds_b128 %0, %1, %2"
                 :: "v"(lds_seg), "v"(byteOff), "s"(seg_map) : "memory");
    asm volatile("global_load_async_to_lds_b128 %0, %1, %2"
                 :: "v"(lds_hf), "v"(byteOff), "s"(height_field) : "memory");
    asm volatile("s_wait_asynccnt 0" ::: "memory");
    __syncthreads();

    const v4i seg4 = s_tile[tid];
    const v4i hf4  = s_tile[THREADS + tid];

    int sv[4], hv[4], rv[4];
#pragma unroll
    for (int c = 0; c < 4; ++c) {
        sv[c] = seg4[c];
        hv[c] = hf4[c];
        const bool is_fp = (sv[c] >= FP_MIN) & (sv[c] < FP_MAX);
        rv[c] = sv[c] + (is_fp ? 1 : 0);   // roof voxel id (z == hf-1)
    }

    const int z0   = (int)blockIdx.y * Z_CHUNK;
    const int pix  = (int)blockIdx.x * (THREADS * 4) + tid * 4;
    int* p = out + (size_t)z0 * HW + pix;

#pragma unroll 4
    for (int i = 0; i < Z_CHUNK; ++i) {
        const int z = z0 + i;
        v4i r;
#pragma unroll
        for (int c = 0; c < 4; ++c) {
            const int v = (z == hv[c] - 1) ? rv[c] : sv[c];
            r[c] = (z < hv[c]) ? v : 0;
        }
        // Streaming output: non-temporal 128-bit store, don't pollute L2.
        __builtin_nontemporal_store(r, (v4i*)p);
        p += HW;
    }
}

extern "C" void kernel_launch(void* const* d_in, const int* in_sizes, int n_in,
                              void* d_out, int out_size, void* d_ws, size_t ws_size,
                              hipStream_t stream) {
    (void)in_sizes; (void)n_in; (void)out_size; (void)d_ws; (void)ws_size;
    const int* height_field = (const int*)d_in[0];
    const int* seg_map      = (const int*)d_in[1];
    int*       out          = (int*)d_out;   // int32 output volume

    dim3 grid(HW / (THREADS * 4), ZMAX / Z_CHUNK);   // (576, 24)
    footprint_extrude_kernel<<<grid, THREADS, 0, stream>>>(height_field, seg_map, out);
}